// WorldModelLoss_60636348285554
// MI455X (gfx1250) — compile-verified
//
#include <hip/hip_runtime.h>
#include <hip/hip_bf16.h>
#include <math.h>

#define TAU_INV (1.0f/0.07f)
#define W_FOCAL 0.6f
#define W_CONTRASTIVE 0.2f
#define IGNORE_INDEX (-100)
#define MAX_ANCHORS 8

#define B_ 64
#define M_ 128
#define V_ 5003
#define S_ 200
#define D_ 128
#define SA_STRIDE 132   // pad 128 -> 132 to avoid LDS bank conflicts

typedef __attribute__((ext_vector_type(2))) float v2f;
typedef __attribute__((ext_vector_type(8))) float v8f;

// ---------------------------------------------------------------- accumulators
__global__ void init_accum(float* acc) {
    if (threadIdx.x < 4) acc[threadIdx.x] = 0.0f;
}

// ------------------------------------------------------- block reduce helpers
// 256 threads = 8 waves. Wave-level shuffle reduce, then 8 partials in LDS.
__device__ __forceinline__ float block_reduce_max(float v, float* s8) {
    for (int m = 1; m < 32; m <<= 1) v = fmaxf(v, __shfl_xor(v, m, 32));
    int wave = threadIdx.x >> 5;
    if ((threadIdx.x & 31) == 0) s8[wave] = v;
    __syncthreads();
    float r = s8[0];
    #pragma unroll
    for (int w = 1; w < 8; w++) r = fmaxf(r, s8[w]);
    __syncthreads();
    return r;
}
__device__ __forceinline__ float block_reduce_sum(float v, float* s8) {
    for (int m = 1; m < 32; m <<= 1) v += __shfl_xor(v, m, 32);
    int wave = threadIdx.x >> 5;
    if ((threadIdx.x & 31) == 0) s8[wave] = v;
    __syncthreads();
    float r = s8[0];
    #pragma unroll
    for (int w = 1; w < 8; w++) r += s8[w];
    __syncthreads();
    return r;
}

// ---------------------------------------------------------------- focal loss
// One block (256 threads) per (b,m) row. Two-pass softmax:
//   pass 1: max over V (pure v_max, b128 loads, fully pipelined from HBM)
//   pass 2: sum exp(v - max) (row is resident in the 192MB L2 by now)
// V=5003 is odd -> scalar head/tail around a 16B-aligned float4 body.
__global__ void __launch_bounds__(256)
focal_kernel(const float* __restrict__ logits, const int* __restrict__ targets,
             const int* __restrict__ mmask, float* __restrict__ acc) {
    int row = blockIdx.x;                 // 0 .. B*M-1
    int t = targets[row];
    bool valid = (t != IGNORE_INDEX) && (mmask[row] == 1);
    if (!valid) return;
    int tt = t < 0 ? 0 : t;

    size_t base = (size_t)row * V_;
    const float* x = logits + base;
    int pad  = (int)((4 - (base & 3)) & 3);   // scalar elements before alignment
    int nvec = (V_ - pad) >> 2;               // aligned float4 count
    int tail = V_ - pad - (nvec << 2);        // scalar elements after
    const float4* xv = (const float4*)(x + pad);

    __shared__ float s8[8];

    // ---- pass 1: max
    float mx = -3.0e38f;
    for (int j = threadIdx.x; j < nvec; j += 256) {
        float4 v = xv[j];
        mx = fmaxf(mx, fmaxf(fmaxf(v.x, v.y), fmaxf(v.z, v.w)));
    }
    if ((int)threadIdx.x < pad)  mx = fmaxf(mx, x[threadIdx.x]);
    if ((int)threadIdx.x < tail) mx = fmaxf(mx, x[pad + (nvec << 2) + threadIdx.x]);
    float MX = block_reduce_max(mx, s8);

    // ---- pass 2: sum of exp(v - MX)
    float sm = 0.0f;
    for (int j = threadIdx.x; j < nvec; j += 256) {
        float4 v = xv[j];
        sm += __expf(v.x - MX) + __expf(v.y - MX) +
              __expf(v.z - MX) + __expf(v.w - MX);
    }
    if ((int)threadIdx.x < pad)  sm += __expf(x[threadIdx.x] - MX);
    if ((int)threadIdx.x < tail) sm += __expf(x[pad + (nvec << 2) + threadIdx.x] - MX);
    float SM = block_reduce_sum(sm, s8);

    if (threadIdx.x == 0) {
        float lse = MX + __logf(SM);
        float lp  = x[tt] - lse;          // log p_t
        float pt  = __expf(lp);
        float om  = 1.0f - pt;
        float fl  = om * om * (-lp);      // gamma == 2
        atomicAdd(&acc[0], fl);
        atomicAdd(&acc[1], 1.0f);
    }
}

// ---------------------------------------------------------------- normalize
// One wave per [B*S] row of D=128; writes normalized embeddings + valid flags.
__global__ void __launch_bounds__(256)
normalize_kernel(const float* __restrict__ emb, const int* __restrict__ pids,
                 const int* __restrict__ amask, float* __restrict__ eN,
                 int* __restrict__ flags) {
    int warp = threadIdx.x >> 5;
    int lane = threadIdx.x & 31;
    int row  = blockIdx.x * 8 + warp;     // < B*S
    if (row >= B_ * S_) return;
    const float4* src = (const float4*)(emb + (size_t)row * D_);
    float4 v = src[lane];
    float ss = v.x*v.x + v.y*v.y + v.z*v.z + v.w*v.w;
    for (int m = 1; m < 32; m <<= 1) ss += __shfl_xor(ss, m, 32);
    float nrm = sqrtf(ss);
    nrm = nrm > 1e-12f ? nrm : 1e-12f;
    float inv = 1.0f / nrm;
    float4 o = make_float4(v.x*inv, v.y*inv, v.z*inv, v.w*inv);
    ((float4*)(eN + (size_t)row * D_))[lane] = o;
    if (lane == 0) flags[row] = (amask[row] == 1 && pids[row] > 0) ? 1 : 0;
}

// ---------------------------------------------------------------- anchor select
// Reproduces pos_sorted / clip semantics: first 16 valid positions (pad S),
// a_idx = clip(vp[0..7], S-1), p_idx = clip(vp[n_anch + 0..7], S-1).
__global__ void select_kernel(const int* __restrict__ flags, int* __restrict__ aidx,
                              int* __restrict__ pidx, int* __restrict__ nanch) {
    int b = threadIdx.x;
    if (b >= B_) return;
    int vp[16];
    #pragma unroll
    for (int i = 0; i < 16; i++) vp[i] = S_;
    int n = 0;
    for (int s = 0; s < S_; s++) {
        if (flags[b * S_ + s]) { if (n < 16) vp[n] = s; n++; }
    }
    int na = n >> 1; if (na > MAX_ANCHORS) na = MAX_ANCHORS;
    nanch[b] = na;
    #pragma unroll
    for (int i = 0; i < 8; i++) {
        int a = vp[i];      if (a > S_ - 1) a = S_ - 1;
        int p = vp[na + i]; if (p > S_ - 1) p = S_ - 1;
        aidx[b * 8 + i] = a;
        pidx[b * 8 + i] = p;
    }
}

// ---------------------------------------------------------------- contrastive
// One wave32 per batch. neg_sim = A(16x128, anchors padded) @ B(128x800)^T via
// chained V_WMMA_F32_16X16X4_F32 (32 k-steps per 16-col tile, 50 tiles),
// fused with an online logsumexp.
__global__ void __launch_bounds__(32)
contrastive_kernel(const float* __restrict__ eN, const int* __restrict__ flags,
                   const int* __restrict__ aidx, const int* __restrict__ pidx,
                   const int* __restrict__ nanch, float* __restrict__ acc) {
    __shared__ float sA[16 * SA_STRIDE];
    __shared__ float sPos[8];
    int b    = blockIdx.x;
    int lane = threadIdx.x;
    int na   = nanch[b];

    // anchors -> LDS rows 0..7; rows 8..15 zero padding
    for (int r = 0; r < 16; ++r) {
        float4 v = make_float4(0.f, 0.f, 0.f, 0.f);
        if (r < 8) {
            int s = aidx[b * 8 + r];
            const float4* src = (const float4*)(eN + ((size_t)(b * S_ + s)) * D_);
            v = src[lane];
        }
        int c = lane * 4;
        sA[r * SA_STRIDE + c + 0] = v.x;
        sA[r * SA_STRIDE + c + 1] = v.y;
        sA[r * SA_STRIDE + c + 2] = v.z;
        sA[r * SA_STRIDE + c + 3] = v.w;
    }
    __syncthreads();
    if (lane < 8) {
        int s = pidx[b * 8 + lane];
        const float* p = eN + ((size_t)(b * S_ + s)) * D_;
        float d = 0.f;
        for (int k = 0; k < D_; k++) d += sA[lane * SA_STRIDE + k] * p[k];
        sPos[lane] = d * TAU_INV;
    }
    __syncthreads();

    float mx[8], sm[8];
    #pragma unroll
    for (int m = 0; m < 8; m++) { mx[m] = -1e30f; sm[m] = 0.f; }

    const int halfsel = (lane >= 16) ? 2 : 0;   // K sub-offset per WMMA layout
    const int nrow16  = lane & 15;              // M (A) / N (B) index

    for (int t = 0; t < (4 * S_) / 16; ++t) {   // 50 tiles of 16 negatives
        int col = t * 16 + nrow16;              // flattened negative index < 800
        int nb  = col / S_;
        int s   = col % S_;
        int nbatch = (nb < b) ? nb : nb + 1;    // first-4 "other batches"
        const float* rowp = eN + ((size_t)(nbatch * S_ + s)) * D_;
        int nvalid = flags[nbatch * S_ + s];

        v8f c = {0.f, 0.f, 0.f, 0.f, 0.f, 0.f, 0.f, 0.f};
        #pragma unroll
        for (int kk = 0; kk < 32; ++kk) {
            int kb = kk * 4 + halfsel;
            v2f a, bb;
            a.x = sA[nrow16 * SA_STRIDE + kb];
            a.y = sA[nrow16 * SA_STRIDE + kb + 1];
            float2 g = *(const float2*)(rowp + kb);
            bb.x = g.x; bb.y = g.y;
            c = __builtin_amdgcn_wmma_f32_16x16x4_f32(
                    false, a, false, bb, (short)0, c, false, false);
        }
        // lanes 0..15: c[m] = sim(anchor m, negative col); lanes 16..31 = padding
        #pragma unroll
        for (int m = 0; m < 8; m++) {
            float v = c[m] * TAU_INV;
            v = nvalid ? v : -1e9f;
            if (v > mx[m]) { sm[m] = sm[m] * __expf(mx[m] - v) + 1.0f; mx[m] = v; }
            else           { sm[m] += __expf(v - mx[m]); }
        }
    }

    // merge (max, sumexp) across lanes 0..15
    #pragma unroll
    for (int m = 0; m < 8; m++) {
        float M = mx[m], Ssum = sm[m];
        for (int off = 1; off < 16; off <<= 1) {
            float M2 = __shfl_xor(M, off, 32);
            float S2 = __shfl_xor(Ssum, off, 32);
            float Mn = M > M2 ? M : M2;
            Ssum = Ssum * __expf(M - Mn) + S2 * __expf(M2 - Mn);
            M = Mn;
        }
        mx[m] = M; sm[m] = Ssum;
    }

    if (lane == 0) {
        float loss = 0.f;
        for (int m = 0; m < na; m++) {
            float ps = sPos[m];
            float M = mx[m], Ssum = sm[m];
            float Mn = M > ps ? M : ps;
            float St = Ssum * __expf(M - Mn) + __expf(ps - Mn);
            float lse = Mn + __logf(St);
            loss += lse - ps;
        }
        if (na > 0) {
            atomicAdd(&acc[2], loss);
            atomicAdd(&acc[3], (float)na);
        }
    }
}

// ---------------------------------------------------------------- finalize
__global__ void finalize_kernel(const float* __restrict__ acc, float* __restrict__ out) {
    float focal = acc[1] > 0.f ? acc[0] / acc[1] : 0.f;
    float closs = acc[3] > 0.f ? acc[2] / acc[3] : 0.f;
    out[0] = W_FOCAL * focal + W_CONTRASTIVE * closs;
}

// ---------------------------------------------------------------- launch
extern "C" void kernel_launch(void* const* d_in, const int* in_sizes, int n_in,
                              void* d_out, int out_size, void* d_ws, size_t ws_size,
                              hipStream_t stream) {
    const float* logits  = (const float*)d_in[0];  // [B,M,V]
    const int*   targets = (const int*)  d_in[1];  // [B,M]
    const int*   mmask   = (const int*)  d_in[2];  // [B,M]
    const float* emb     = (const float*)d_in[3];  // [B,S,D]
    const int*   pids    = (const int*)  d_in[4];  // [B,S]
    const int*   amask   = (const int*)  d_in[5];  // [B,S]
    float* out = (float*)d_out;

    // workspace layout
    float* acc   = (float*)d_ws;                       // [0]=fsum [1]=fcnt [2]=csum [3]=pairs
    float* eN    = acc + 16;                           // B*S*D normalized embeddings
    int*   flags = (int*)(eN + (size_t)B_ * S_ * D_);  // B*S
    int*   aidx  = flags + B_ * S_;                    // B*8
    int*   pidx  = aidx + B_ * 8;                      // B*8
    int*   nanch = pidx + B_ * 8;                      // B

    hipLaunchKernelGGL(init_accum,        dim3(1),             dim3(32),  0, stream, acc);
    hipLaunchKernelGGL(normalize_kernel,  dim3((B_ * S_) / 8), dim3(256), 0, stream,
                       emb, pids, amask, eN, flags);
    hipLaunchKernelGGL(select_kernel,     dim3(1),             dim3(64),  0, stream,
                       flags, aidx, pidx, nanch);
    hipLaunchKernelGGL(focal_kernel,      dim3(B_ * M_),       dim3(256), 0, stream,
                       logits, targets, mmask, acc);
    hipLaunchKernelGGL(contrastive_kernel, dim3(B_),           dim3(32),  0, stream,
                       eN, flags, aidx, pidx, nanch, acc);
    hipLaunchKernelGGL(finalize_kernel,   dim3(1),             dim3(1),   0, stream, acc, out);
}